// DualPromptNAFNetv3_32676111188195
// MI455X (gfx1250) — compile-verified
//
#include <hip/hip_runtime.h>
#include <math.h>

typedef __attribute__((ext_vector_type(2))) float v2f;
typedef __attribute__((ext_vector_type(8))) float v8f;

constexpr int Bn = 16384;
constexpr int Pn = 15;
constexpr int Dn = 1024;
constexpr int TDn = 5 * 1024;   // T*D = 5120

// ---------------------------------------------------------------------------
// Kernel 1: knn[p] = prompt_keys[p] / (max(||k_p||, eps) * clamp(temp,0.1,2))
// Row 15 zero-padded so the WMMA B tile (N=16) reads zeros for the fake col.
// ---------------------------------------------------------------------------
__global__ __launch_bounds__(256) void k_norm_keys(const float* __restrict__ pk,
                                                   const float* __restrict__ temp,
                                                   float* __restrict__ knn) {
  __shared__ float red[256];
  __shared__ float scale;
  const int p = blockIdx.x, tid = threadIdx.x;
  if (p == Pn) {  // pad row
    for (int i = tid; i < Dn; i += 256) knn[Pn * Dn + i] = 0.f;
    return;
  }
  float ss = 0.f;
  for (int i = tid; i < Dn; i += 256) { float v = pk[p * Dn + i]; ss += v * v; }
  red[tid] = ss;
  __syncthreads();
  for (int s = 128; s > 0; s >>= 1) {
    if (tid < s) red[tid] += red[tid + s];
    __syncthreads();
  }
  if (tid == 0) {
    float t = fminf(fmaxf(temp[0], 0.1f), 2.0f);
    scale = 1.f / (fmaxf(sqrtf(red[0]), 1e-12f) * t);
  }
  __syncthreads();
  const float sc = scale;
  for (int i = tid; i < Dn; i += 256) knn[p * Dn + i] = pk[p * Dn + i] * sc;
}

// ---------------------------------------------------------------------------
// Kernel 2: pre[p][d] = sum_j agg_w[d][j] * PV_flat[p][j]   (15 x 1024 out)
// Tiny (157 MFLOP); agg_w streamed once (20 MB), PV stays in L2.
// ---------------------------------------------------------------------------
__global__ __launch_bounds__(256) void k_premix(const float* __restrict__ agg_w,
                                                const float* __restrict__ pv,
                                                float* __restrict__ pre) {
  const int tid = threadIdx.x;
  const int d = blockIdx.x * 16 + (tid & 15);
  const int p = tid >> 4;
  if (p >= Pn) return;
  const float4* wr = reinterpret_cast<const float4*>(agg_w + (size_t)d * TDn);
  const float4* vr = reinterpret_cast<const float4*>(pv + (size_t)p * TDn);
  float acc = 0.f;
#pragma unroll 4
  for (int j = 0; j < TDn / 4; ++j) {
    float4 a = wr[j], b = vr[j];
    acc = fmaf(a.x, b.x, acc);
    acc = fmaf(a.y, b.y, acc);
    acc = fmaf(a.z, b.z, acc);
    acc = fmaf(a.w, b.w, acc);
  }
  pre[p * Dn + d] = acc;
}

// ---------------------------------------------------------------------------
// Kernel 3: fused sim (f32 WMMA) -> top3 -> softmax -> pre-combine -> LN -> GELU
// 128 threads = 4 waves; each wave owns 16 rows of B. Grid = 16384/64 = 256.
// ---------------------------------------------------------------------------
__device__ __forceinline__ float wave_sum(float v) {
  v += __shfl_xor(v, 1, 32);
  v += __shfl_xor(v, 2, 32);
  v += __shfl_xor(v, 4, 32);
  v += __shfl_xor(v, 8, 32);
  v += __shfl_xor(v, 16, 32);
  return v;
}

__global__ __launch_bounds__(128) void k_main(const float* __restrict__ q,
                                              const float* __restrict__ knn,
                                              const float* __restrict__ pre,
                                              const float* __restrict__ bias,
                                              const float* __restrict__ lnw,
                                              const float* __restrict__ lnb,
                                              float* __restrict__ out,
                                              int* __restrict__ idx_out) {
  __shared__ __align__(16) float preS[Pn * Dn];     // 60 KB, block-resident
  __shared__ __align__(16) float biasS[Dn];
  __shared__ __align__(16) float lnwS[Dn];
  __shared__ __align__(16) float lnbS[Dn];
  __shared__ float simS[4][16][17];
  __shared__ float twS[4][16][3];
  __shared__ int   tiS[4][16][3];

  const int tid = threadIdx.x, wave = tid >> 5, lane = tid & 31;
  const int lo = lane & 15, hi = lane >> 4;

  for (int i = tid; i < Pn * Dn; i += 128) preS[i] = pre[i];
  for (int i = tid; i < Dn; i += 128) {
    biasS[i] = bias[i];
    lnwS[i] = lnw[i];
    lnbS[i] = lnb[i];
  }

  const int b0 = blockIdx.x * 64 + wave * 16;

  // ---- phase 1: 16x16 sim tile via V_WMMA_F32_16X16X4_F32, fused row norms --
  // A frag (16x4 f32): lane holds q[b0+lo][k0 + 2*hi + {0,1}]
  // B frag (4x16 f32): lane holds knn[lo][k0 + 2*hi + {0,1}]  (B[k][n]=knn[n][k])
  const float* aptr = q + (size_t)(b0 + lo) * Dn + 2 * hi;
  const float* bptr = knn + (size_t)lo * Dn + 2 * hi;
  v8f c = {0.f, 0.f, 0.f, 0.f, 0.f, 0.f, 0.f, 0.f};
  float ss = 0.f;
#pragma unroll 4
  for (int k0 = 0; k0 < Dn; k0 += 4) {
    v2f a = *reinterpret_cast<const v2f*>(aptr + k0);
    v2f b = *reinterpret_cast<const v2f*>(bptr + k0);
    ss = fmaf(a.x, a.x, ss);
    ss = fmaf(a.y, a.y, ss);
    c = __builtin_amdgcn_wmma_f32_16x16x4_f32(false, a, false, b, (short)0, c,
                                              false, false);
  }
  // lanes (l, l+16) each saw half of row lo -> combine
  ss += __shfl_xor(ss, 16, 32);
  const float denom = fmaxf(sqrtf(ss), 1e-12f);  // ||q_row||, temp folded in knn
#pragma unroll
  for (int r = 0; r < 8; ++r) {
    const int M = r + 8 * hi;                    // C layout: VGPR r, half `hi`
    const float dM = __shfl(denom, M, 32);
    simS[wave][M][lo] = c[r] / dM;
  }
  __syncthreads();

  // ---- phase 2: per-row top-3 (torch tie-break: strict >) + softmax --------
  if (lane < 16) {
    const int row = lane, b = b0 + row;
    float s0 = -1e30f, s1 = -1e30f, s2 = -1e30f;
    int i0 = 0, i1 = 0, i2 = 0;
#pragma unroll
    for (int p = 0; p < Pn; ++p) {
      const float v = simS[wave][row][p];
      if (v > s0)      { s2 = s1; i2 = i1; s1 = s0; i1 = i0; s0 = v; i0 = p; }
      else if (v > s1) { s2 = s1; i2 = i1; s1 = v; i1 = p; }
      else if (v > s2) { s2 = v; i2 = p; }
    }
    const float e1 = expf(s1 - s0), e2 = expf(s2 - s0);
    const float inv = 1.f / (1.f + e1 + e2);
    twS[wave][row][0] = inv;
    twS[wave][row][1] = e1 * inv;
    twS[wave][row][2] = e2 * inv;
    tiS[wave][row][0] = i0;
    tiS[wave][row][1] = i1;
    tiS[wave][row][2] = i2;
    idx_out[b * 3 + 0] = i0;
    idx_out[b * 3 + 1] = i1;
    idx_out[b * 3 + 2] = i2;
  }
  __syncthreads();

  // ---- phase 3: lin = Σ w_k*pre[i_k] + bias; LayerNorm; exact erf-GELU -----
  for (int r = 0; r < 16; ++r) {
    const int b = b0 + r;
    const float w0 = twS[wave][r][0], w1 = twS[wave][r][1], w2 = twS[wave][r][2];
    const float* p0 = preS + tiS[wave][r][0] * Dn;
    const float* p1 = preS + tiS[wave][r][1] * Dn;
    const float* p2 = preS + tiS[wave][r][2] * Dn;
    float4 lin[8];
    float sum = 0.f;
#pragma unroll
    for (int j = 0; j < 8; ++j) {
      const int d = (j * 32 + lane) * 4;
      float4 a0 = *reinterpret_cast<const float4*>(p0 + d);
      float4 a1 = *reinterpret_cast<const float4*>(p1 + d);
      float4 a2 = *reinterpret_cast<const float4*>(p2 + d);
      float4 bs = *reinterpret_cast<const float4*>(biasS + d);
      float4 v;
      v.x = fmaf(w0, a0.x, fmaf(w1, a1.x, fmaf(w2, a2.x, bs.x)));
      v.y = fmaf(w0, a0.y, fmaf(w1, a1.y, fmaf(w2, a2.y, bs.y)));
      v.z = fmaf(w0, a0.z, fmaf(w1, a1.z, fmaf(w2, a2.z, bs.z)));
      v.w = fmaf(w0, a0.w, fmaf(w1, a1.w, fmaf(w2, a2.w, bs.w)));
      lin[j] = v;
      sum += v.x + v.y + v.z + v.w;
    }
    const float mu = wave_sum(sum) * (1.f / Dn);
    float vs = 0.f;
#pragma unroll
    for (int j = 0; j < 8; ++j) {
      float dx;
      dx = lin[j].x - mu; vs = fmaf(dx, dx, vs);
      dx = lin[j].y - mu; vs = fmaf(dx, dx, vs);
      dx = lin[j].z - mu; vs = fmaf(dx, dx, vs);
      dx = lin[j].w - mu; vs = fmaf(dx, dx, vs);
    }
    const float var = wave_sum(vs) * (1.f / Dn);
    const float istd = rsqrtf(var + 1e-5f);
#pragma unroll
    for (int j = 0; j < 8; ++j) {
      const int d = (j * 32 + lane) * 4;
      float4 gw = *reinterpret_cast<const float4*>(lnwS + d);
      float4 gb = *reinterpret_cast<const float4*>(lnbS + d);
      float4 o;
      float y;
      y = (lin[j].x - mu) * istd * gw.x + gb.x;
      o.x = 0.5f * y * (1.f + erff(y * 0.70710678118654752440f));
      y = (lin[j].y - mu) * istd * gw.y + gb.y;
      o.y = 0.5f * y * (1.f + erff(y * 0.70710678118654752440f));
      y = (lin[j].z - mu) * istd * gw.z + gb.z;
      o.z = 0.5f * y * (1.f + erff(y * 0.70710678118654752440f));
      y = (lin[j].w - mu) * istd * gw.w + gb.w;
      o.w = 0.5f * y * (1.f + erff(y * 0.70710678118654752440f));
      *reinterpret_cast<float4*>(out + (size_t)b * Dn + d) = o;
    }
  }
}

// ---------------------------------------------------------------------------
extern "C" void kernel_launch(void* const* d_in, const int* in_sizes, int n_in,
                              void* d_out, int out_size, void* d_ws, size_t ws_size,
                              hipStream_t stream) {
  const float* q     = (const float*)d_in[0];  // [B, D]
  const float* pk    = (const float*)d_in[1];  // [P, D]
  const float* pv    = (const float*)d_in[2];  // [P, T, D]
  const float* temp  = (const float*)d_in[3];  // [1]
  const float* agg_w = (const float*)d_in[4];  // [D, T*D]
  const float* agg_b = (const float*)d_in[5];  // [D]
  const float* ln_w  = (const float*)d_in[6];  // [D]
  const float* ln_b  = (const float*)d_in[7];  // [D]

  float* knn = (float*)d_ws;                   // 16*1024 floats (row 15 = pad)
  float* pre = knn + 16 * Dn;                  // 15*1024 floats

  float* outp = (float*)d_out;                             // [B, D] f32
  int*   idxp = (int*)(outp + (size_t)Bn * Dn);            // [B, 3] i32

  k_norm_keys<<<Pn + 1, 256, 0, stream>>>(pk, temp, knn);
  k_premix<<<Dn / 16, 256, 0, stream>>>(agg_w, pv, pre);
  k_main<<<Bn / 64, 128, 0, stream>>>(q, knn, pre, agg_b, ln_w, ln_b, outp, idxp);
}